// Attention_65335042507066
// MI455X (gfx1250) — compile-verified
//
#include <hip/hip_runtime.h>

// fp32 matvec  out[S] = A[S,H] @ x[H]  via V_WMMA_F32_16X16X4_F32 with
// B-broadcast (every column of B = x-chunk), K-split x16 for occupancy,
// native f32 atomics for the partial reduction.
// Roofline: 256 MB of A streamed once ~= 11 us at 23.3 TB/s; compute is
// negligible, so A loads are non-temporal (A > L2) and x stays temporal.

typedef __attribute__((ext_vector_type(2))) float v2f;
typedef __attribute__((ext_vector_type(8))) float v8f;

constexpr int KSPLIT = 16;  // waves per 16-row tile (each does H/KSPLIT cols)

__global__ __launch_bounds__(256) void matvec_wmma16x16x4(
    const float* __restrict__ A, const float* __restrict__ x,
    float* __restrict__ out, int S, int H)
{
    const int KLEN  = H / KSPLIT;                       // 256 for H=4096
    const int wave  = (int)((blockIdx.x * blockDim.x + threadIdx.x) >> 5);
    const int lane  = (int)(threadIdx.x & 31);
    const int tile  = wave / KSPLIT;
    const int chunk = wave % KSPLIT;
    if (tile >= (S >> 4)) return;                       // wave-uniform guard

    const int r0    = tile << 4;
    const int hi    = lane >> 4;                        // 0: K={0,1}, 1: K={2,3}
    const int m     = lane & 15;                        // row within tile (A side)
    const int hioff = hi ? 2 : 0;

    // A fragment lane address: row (r0+m), cols kbase + {0,1} or {2,3}
    const float* pA = A + (size_t)(r0 + m) * (size_t)H
                        + (size_t)(chunk * KLEN) + hioff;
    // B fragment: same K striping, broadcast across all 16 columns -> same
    // float2 of x for every lane in a half-wave. x is 16 KB, L2/L0 resident.
    const float* pX = x + chunk * KLEN + hioff;

    v8f acc0 = {};
    v8f acc1 = {};   // two accumulators to break WMMA->WMMA RAW chains

    for (int k = 0; k < KLEN; k += 16) {
        if (k + 80 <= KLEN)
            __builtin_prefetch(pA + k + 64, 0, 3);      // WGP-scope prefetch

        // A: streamed once -> non-temporal (don't displace L2-resident x)
        v2f a0 = __builtin_nontemporal_load((const v2f*)(pA + k));
        v2f b0 = *(const v2f*)(pX + k);
        v2f a1 = __builtin_nontemporal_load((const v2f*)(pA + k + 4));
        v2f b1 = *(const v2f*)(pX + k + 4);
        v2f a2 = __builtin_nontemporal_load((const v2f*)(pA + k + 8));
        v2f b2 = *(const v2f*)(pX + k + 8);
        v2f a3 = __builtin_nontemporal_load((const v2f*)(pA + k + 12));
        v2f b3 = *(const v2f*)(pX + k + 12);

        acc0 = __builtin_amdgcn_wmma_f32_16x16x4_f32(
                   false, a0, false, b0, (short)0, acc0, false, false);
        acc1 = __builtin_amdgcn_wmma_f32_16x16x4_f32(
                   false, a1, false, b1, (short)0, acc1, false, false);
        acc0 = __builtin_amdgcn_wmma_f32_16x16x4_f32(
                   false, a2, false, b2, (short)0, acc0, false, false);
        acc1 = __builtin_amdgcn_wmma_f32_16x16x4_f32(
                   false, a3, false, b3, (short)0, acc1, false, false);
    }

    v8f acc = acc0 + acc1;

    // D layout: lanes 0-15 hold rows 0..7 in VGPRs 0..7 (all columns equal
    // because B was broadcast); lanes 16-31 hold rows 8..15. One lane per
    // half-wave commits its 8 partial sums.
    if (m == 0) {
        float* po = out + r0 + (hi ? 8 : 0);
#pragma unroll
        for (int i = 0; i < 8; ++i)
            unsafeAtomicAdd(po + i, acc[i]);            // global_atomic_add_f32
    }
}

extern "C" void kernel_launch(void* const* d_in, const int* in_sizes, int n_in,
                              void* d_out, int out_size, void* d_ws, size_t ws_size,
                              hipStream_t stream)
{
    const float* A   = (const float*)d_in[0];           // encoder_out [S,H]
    const float* dec = (const float*)d_in[1];           // decoder_hidden [L,H]
    const int H = 4096;
    const int S = in_sizes[0] / H;                      // 16384
    const float* x = dec + (in_sizes[1] - H);           // top layer = last row
    float* out = (float*)d_out;

    // Zero output (harness poisons it); atomics accumulate partials into it.
    hipMemsetAsync(out, 0, (size_t)S * sizeof(float), stream);

    const int tiles   = S / 16;                         // 1024
    const int waves   = tiles * KSPLIT;                 // 16384 waves
    const int block   = 256;                            // 8 waves / block
    const int threads = waves * 32;
    const int grid    = (threads + block - 1) / block;  // 2048 blocks

    matvec_wmma16x16x4<<<grid, block, 0, stream>>>(A, x, out, S, H);
}